// SplineGNN_49289044689245
// MI455X (gfx1250) — compile-verified
//
#include <hip/hip_runtime.h>
#include <math.h>

typedef float v2f __attribute__((ext_vector_type(2)));
typedef float v8f __attribute__((ext_vector_type(8)));

// ---------------------------------------------------------------------------
// Build concatenated weight matrices
//   Wc1[192][64] = [ W1[0] ; W1[1]-W1[0] ; root1 ]
//   Wc2[192][64] = [ W2[0] ; W2[1]-W2[0] ; root2 ] zero-padded from 40 -> 64 cols
// ---------------------------------------------------------------------------
__global__ void prep_weights_k(const float* __restrict__ W1, const float* __restrict__ root1,
                               const float* __restrict__ W2, const float* __restrict__ root2,
                               float* __restrict__ Wc1, float* __restrict__ Wc2) {
  int i = blockIdx.x * blockDim.x + threadIdx.x;
  if (i >= 192 * 64) return;
  int k = i >> 6, c = i & 63;
  float v1;
  if (k < 64)       v1 = W1[k * 64 + c];
  else if (k < 128) v1 = W1[4096 + (k - 64) * 64 + c] - W1[(k - 64) * 64 + c];
  else              v1 = root1[(k - 128) * 64 + c];
  Wc1[i] = v1;
  float v2 = 0.0f;
  if (c < 40) {
    if (k < 64)       v2 = W2[k * 40 + c];
    else if (k < 128) v2 = W2[2560 + (k - 64) * 40 + c] - W2[(k - 64) * 40 + c];
    else              v2 = root2[(k - 128) * 40 + c];
  }
  Wc2[i] = v2;
}

// ---------------------------------------------------------------------------
// CSR build, stage 1: degree histogram (int atomics only)
// ---------------------------------------------------------------------------
__global__ void count_k(const int* __restrict__ dst, int* __restrict__ deg, int E) {
  int e = blockIdx.x * blockDim.x + threadIdx.x;
  if (e < E) atomicAdd(&deg[dst[e]], 1);
}

// ---------------------------------------------------------------------------
// CSR build, stage 2: exclusive prefix sum over deg[0..N) -> rowptr[0..N].
// Single workgroup of 1024 threads (32 wave32s); wave scans via shfl_up,
// cross-wave combine via LDS. Also initializes fill[] = rowptr[].
// ---------------------------------------------------------------------------
__global__ __launch_bounds__(1024) void scan_k(const int* __restrict__ deg,
                                               int* __restrict__ rowptr,
                                               int* __restrict__ fill, int Nn) {
  __shared__ int wsum[32];
  __shared__ int carrySh;
  int tid = threadIdx.x;
  int lane = tid & 31;
  int wid = tid >> 5;
  if (tid == 0) carrySh = 0;
  __syncthreads();
  for (int base = 0; base < Nn; base += 1024) {
    int i = base + tid;
    int v = (i < Nn) ? deg[i] : 0;
    int incl = v;
    for (int off = 1; off < 32; off <<= 1) {
      int t = __shfl_up(incl, off, 32);
      if (lane >= off) incl += t;
    }
    if (lane == 31) wsum[wid] = incl;
    __syncthreads();
    if (wid == 0) {
      int winc = wsum[lane];
      for (int off = 1; off < 32; off <<= 1) {
        int t = __shfl_up(winc, off, 32);
        if (lane >= off) winc += t;
      }
      wsum[lane] = winc;                    // inclusive scan of wave sums
    }
    __syncthreads();
    int waveOff = (wid == 0) ? 0 : wsum[wid - 1];
    int carry = carrySh;
    int excl = carry + waveOff + incl - v;
    if (i < Nn) { rowptr[i] = excl; fill[i] = excl; }
    __syncthreads();                         // everyone done reading carrySh/wsum
    if (tid == 1023) carrySh = carry + wsum[31];
    __syncthreads();
  }
  if (tid == 0) rowptr[Nn] = carrySh;
}

// ---------------------------------------------------------------------------
// CSR build, stage 3: bucket-fill edge data (src index + pseudo-coordinate u)
// ---------------------------------------------------------------------------
__global__ void fill_k(const int* __restrict__ src, const int* __restrict__ dst,
                       const float* __restrict__ u, int* __restrict__ fill,
                       int* __restrict__ srcs, float* __restrict__ us, int E) {
  int e = blockIdx.x * blockDim.x + threadIdx.x;
  if (e >= E) return;
  int p = atomicAdd(&fill[dst[e]], 1);
  srcs[p] = src[e];
  us[p] = u[e];
}

// ---------------------------------------------------------------------------
// Per-node aggregation (gather form; NO feature atomics).
// One wave per node; lane holds 2 of 64 features (float2).
//   G[n][  0.. 63] = (1/max(deg,1)) * sum_e x[src_e]
//   G[n][ 64..127] = (1/max(deg,1)) * sum_e u_e * x[src_e]
//   G[n][128..191] = Xsrc[n]                       (root term input)
// ---------------------------------------------------------------------------
__global__ void aggregate_k(const float* __restrict__ Xsrc,
                            const int* __restrict__ rowptr, const int* __restrict__ srcs,
                            const float* __restrict__ us,
                            float* __restrict__ G, int Nn) {
  int n = blockIdx.x * (blockDim.x >> 5) + (threadIdx.x >> 5);
  int lane = threadIdx.x & 31;
  if (n >= Nn) return;
  int p0 = rowptr[n], p1 = rowptr[n + 1];
  float ax = 0.f, ay = 0.f, bx = 0.f, by = 0.f;
  const float2* X2 = (const float2*)Xsrc;
  for (int p = p0; p < p1; ++p) {
    int s = srcs[p];                 // wave-uniform broadcast load (L1 hit)
    float uu = us[p];
    float2 xv = X2[(size_t)s * 32 + lane];   // 256B coalesced row gather (L2)
    ax += xv.x; ay += xv.y;
    bx += uu * xv.x; by += uu * xv.y;
  }
  float ic = 1.0f / fmaxf((float)(p1 - p0), 1.0f);
  float* g = G + (size_t)n * 192;
  float2 va; va.x = ax * ic; va.y = ay * ic;
  float2 vb; vb.x = bx * ic; vb.y = by * ic;
  ((float2*)g)[lane] = va;
  ((float2*)(g + 64))[lane] = vb;
  ((float2*)(g + 128))[lane] = X2[(size_t)n * 32 + lane];
}

// ---------------------------------------------------------------------------
// Node-level GEMM via V_WMMA_F32_16X16X4_F32:  out = G[N][192] @ Wc[192][64].
// A-fragments: pipelined global_load_b64 (row stride 192 floats).
// B-fragments: pair-packed LDS — sW[kp][col] = {Wc[2kp][col], Wc[2kp+1][col]}
//              so one ds_load_b64 lands the fragment in an aligned VGPR pair
//              (no v_dual_mov repacking in the WMMA loop).
// Each wave computes a 16x64 tile (4 accumulators of 16x16).
// mode 0: out = ELU(acc + bias)   (layer 1 -> h)
// mode 1: out = acc + bias        (layer 2 -> logits; bias valid col<nbias)
// Fragment layouts per ISA 7.12.2.
// ---------------------------------------------------------------------------
__global__ __launch_bounds__(128) void node_gemm_k(
    const float* __restrict__ G, const float* __restrict__ Wc,
    const float* __restrict__ bias, float* __restrict__ out,
    int Nn, int mode, int nbias) {
  __shared__ float2 sW[96 * 64];   // 48KB, pair-packed along K
  for (int i = threadIdx.x; i < 96 * 64; i += blockDim.x) {
    int kp = i >> 6, c = i & 63;
    float2 p;
    p.x = Wc[(2 * kp) * 64 + c];
    p.y = Wc[(2 * kp + 1) * 64 + c];
    sW[i] = p;
  }
  __syncthreads();

  int w = blockIdx.x * (blockDim.x >> 5) + (threadIdx.x >> 5);
  int row0 = w * 16;
  if (row0 >= Nn) return;            // wave-uniform: EXEC all-ones for WMMA
  int lane = threadIdx.x & 31;
  int half = lane >> 4;
  int ml = lane & 15;
  int r = row0 + ml;

  v8f zero = {0.f, 0.f, 0.f, 0.f, 0.f, 0.f, 0.f, 0.f};
  v8f acc[4] = {zero, zero, zero, zero};

  const float2* G2 = (const float2*)G;
  for (int kk = 0; kk < 192; kk += 4) {
    int kp = (kk >> 1) + half;                  // this lane's K pair index
    v2f a = __builtin_bit_cast(v2f, G2[(size_t)r * 96 + kp]);
    const float2* wrow = &sW[kp * 64 + ml];
#pragma unroll
    for (int t = 0; t < 4; ++t) {
      v2f b = __builtin_bit_cast(v2f, wrow[t * 16]);   // single ds_load_b64
      acc[t] = __builtin_amdgcn_wmma_f32_16x16x4_f32(
          false, a, false, b, (short)0, acc[t], false, false);
    }
  }

#pragma unroll
  for (int t = 0; t < 4; ++t) {
    int col = t * 16 + ml;
    float bv = (col < nbias) ? bias[col] : 0.0f;
#pragma unroll
    for (int v = 0; v < 8; ++v) {
      int row = row0 + v + 8 * half;
      float val = acc[t][v] + bv;
      if (mode == 0) val = (val > 0.0f) ? val : expm1f(val);  // ELU(alpha=1)
      out[(size_t)row * 64 + col] = val;
    }
  }
}

// ---------------------------------------------------------------------------
// log_softmax over 40 classes; one wave per row (wave32 shfl reductions).
// ---------------------------------------------------------------------------
__global__ void logsoftmax_k(const float* __restrict__ logits, float* __restrict__ out, int Nn) {
  int w = blockIdx.x * (blockDim.x >> 5) + (threadIdx.x >> 5);
  int lane = threadIdx.x & 31;
  if (w >= Nn) return;
  const float* rp = logits + (size_t)w * 64;
  float v0 = rp[lane];
  float v1 = (lane < 8) ? rp[lane + 32] : -INFINITY;
  float m = fmaxf(v0, v1);
  for (int off = 16; off > 0; off >>= 1) m = fmaxf(m, __shfl_xor(m, off, 32));
  float s = expf(v0 - m) + ((lane < 8) ? expf(v1 - m) : 0.0f);
  for (int off = 16; off > 0; off >>= 1) s += __shfl_xor(s, off, 32);
  float ls = m + logf(s);
  out[(size_t)w * 40 + lane] = v0 - ls;
  if (lane < 8) out[(size_t)w * 40 + lane + 32] = v1 - ls;
}

// ---------------------------------------------------------------------------
extern "C" void kernel_launch(void* const* d_in, const int* in_sizes, int n_in,
                              void* d_out, int out_size, void* d_ws, size_t ws_size,
                              hipStream_t stream) {
  const float* x     = (const float*)d_in[0];
  const int*   ei    = (const int*)d_in[1];
  const float* ea    = (const float*)d_in[2];
  const float* W1    = (const float*)d_in[3];
  const float* root1 = (const float*)d_in[4];
  const float* b1    = (const float*)d_in[5];
  const float* W2    = (const float*)d_in[6];
  const float* root2 = (const float*)d_in[7];
  const float* b2    = (const float*)d_in[8];
  float* out = (float*)d_out;

  int Nn = in_sizes[0] / 64;      // 100000
  int E  = in_sizes[2];           // 1600000
  const int* src = ei;
  const int* dst = ei + E;

  size_t N64  = (size_t)Nn * 64;
  size_t N192 = (size_t)Nn * 192;
  size_t Npad = ((size_t)Nn + 63) & ~(size_t)63;

  float* ws     = (float*)d_ws;
  float* G      = ws;                       // N*192
  float* h      = G + N192;                 // N*64
  float* logits = h + N64;                  // N*64
  float* Wc1    = logits + N64;             // 192*64
  float* Wc2    = Wc1 + 192 * 64;           // 192*64
  float* usbuf  = Wc2 + 192 * 64;           // E floats (CSR u values)
  int*   deg    = (int*)(usbuf + E);        // N ints
  int*   rowptr = deg + Npad;               // N+1 ints
  int*   fill   = rowptr + Npad + 64;       // N ints
  int*   srcs   = fill + Npad;              // E ints (CSR src indices)

  // ---- CSR build (once, reused by both layers) ----
  hipMemsetAsync(deg, 0, Npad * sizeof(int), stream);
  prep_weights_k<<<(192 * 64 + 255) / 256, 256, 0, stream>>>(W1, root1, W2, root2, Wc1, Wc2);
  count_k<<<(E + 255) / 256, 256, 0, stream>>>(dst, deg, E);
  scan_k<<<1, 1024, 0, stream>>>(deg, rowptr, fill, Nn);
  fill_k<<<(E + 255) / 256, 256, 0, stream>>>(src, dst, ea, fill, srcs, usbuf, E);

  int ablocks = (Nn + 7) / 8;               // 8 waves (nodes) per 256-thread block
  int tiles   = (Nn + 15) / 16;
  int gblocks = (tiles + 3) / 4;            // 4 waves (tiles) per 128-thread block

  // ---- layer 1 ----
  aggregate_k<<<ablocks, 256, 0, stream>>>(x, rowptr, srcs, usbuf, G, Nn);
  node_gemm_k<<<gblocks, 128, 0, stream>>>(G, Wc1, b1, h, Nn, 0, 64);

  // ---- layer 2 ----
  aggregate_k<<<ablocks, 256, 0, stream>>>(h, rowptr, srcs, usbuf, G, Nn);
  node_gemm_k<<<gblocks, 128, 0, stream>>>(G, Wc2, b2, logits, Nn, 1, 40);

  // ---- output ----
  logsoftmax_k<<<(Nn + 7) / 8, 256, 0, stream>>>(logits, out, Nn);
}